// EuclideanPoolDecoder_72980084294073
// MI455X (gfx1250) — compile-verified
//
#include <hip/hip_runtime.h>
#include <hip/hip_bf16.h>

#define N_NODES   100000
#define N_EDGES   3200000
#define DIM       256
#define N_CLASSES 16
#define N_GRAPHS  1000

typedef __attribute__((ext_vector_type(2))) float v2f;
typedef __attribute__((ext_vector_type(8))) float v8f;

// LDS-transposed W with padded stride to avoid bank conflicts.
#define WT_STRIDE (DIM + 4)   // 260 floats; row starts stay 8B-aligned

// ---------------------------------------------------------------------------
// Kernel 1: hidden[M,16] = x[M,256] @ W[256,16] + b  via V_WMMA_F32_16X16X4_F32
// 256 threads = 8 wave32; each wave produces a 16x16 output tile.
// ---------------------------------------------------------------------------
__global__ __launch_bounds__(256) void gcn_gemm_bias(const float* __restrict__ x,
                                                     const float* __restrict__ W,
                                                     const float* __restrict__ bias,
                                                     float* __restrict__ hidden) {
  __shared__ float Wt[N_CLASSES * WT_STRIDE];  // Wt[n][k] = W[k*16 + n]
  for (int i = threadIdx.x; i < DIM * N_CLASSES; i += 256) {
    int k = i >> 4, n = i & 15;
    Wt[n * WT_STRIDE + k] = W[i];
  }
  __syncthreads();

  const int wave = threadIdx.x >> 5;
  const int lane = threadIdx.x & 31;
  const int half = lane >> 4;       // 0: lanes 0-15, 1: lanes 16-31
  const int l16  = lane & 15;
  const int m_base = blockIdx.x * 128 + wave * 16;
  if (m_base >= N_NODES) return;    // wave-uniform: N_NODES % 16 == 0, EXEC stays full

  // A fragment: lane l16 holds row (m_base+l16); K pair offset 2*half.
  const float* arow = x  + (size_t)(m_base + l16) * DIM + half * 2;
  // B fragment: lane l16 holds column l16; same K pairing as A.
  const float* brow = Wt + l16 * WT_STRIDE + half * 2;

  v8f c0 = {};
  v8f c1 = {};
#pragma unroll 4
  for (int k = 0; k < DIM; k += 8) {
    v2f a0 = *(const v2f*)(arow + k);
    v2f b0 = *(const v2f*)(brow + k);
    c0 = __builtin_amdgcn_wmma_f32_16x16x4_f32(false, a0, false, b0, (short)0, c0, false, false);
    v2f a1 = *(const v2f*)(arow + k + 4);
    v2f b1 = *(const v2f*)(brow + k + 4);
    c1 = __builtin_amdgcn_wmma_f32_16x16x4_f32(false, a1, false, b1, (short)0, c1, false, false);
  }

  const float bn = bias[l16];
#pragma unroll
  for (int j = 0; j < 8; ++j) {
    int row = m_base + j + half * 8;          // C layout: VGPR j -> M=j (lo) / j+8 (hi)
    hidden[(size_t)row * N_CLASSES + l16] = c0[j] + c1[j] + bn;
  }
}

// ---------------------------------------------------------------------------
// Kernel 0: zero the support accumulator in workspace.
// ---------------------------------------------------------------------------
__global__ void gcn_zero(float* __restrict__ p, long long n) {
  long long i = (long long)blockIdx.x * blockDim.x + threadIdx.x;
  long long stride = (long long)gridDim.x * blockDim.x;
  for (; i < n; i += stride) p[i] = 0.0f;
}

// ---------------------------------------------------------------------------
// Device-scope non-returning fp32 atomic add, guaranteed native
// GLOBAL_ATOMIC_ADD_F32 (avoids clang's conservative CAS-loop expansion for
// generic-pointer fp atomics). Tracked with STOREcnt; s_endpgm's implicit
// wait-idle drains it.
// ---------------------------------------------------------------------------
__device__ __forceinline__ void atomic_add_f32_dev(float* addr, float v) {
  asm volatile("global_atomic_add_f32 %0, %1, off scope:SCOPE_DEV"
               :
               : "v"(addr), "v"(v)
               : "memory");
}

// ---------------------------------------------------------------------------
// Kernel 2: COO SpMM scatter: support[rows[e]] += vals[e] * hidden[cols[e]]
// One thread per (edge, channel); 16 lanes share an edge (same-address loads
// coalesce to a broadcast; the 16 atomics land on one contiguous 64B row).
// ---------------------------------------------------------------------------
__global__ __launch_bounds__(256) void gcn_edge_scatter(const long long* __restrict__ rows,
                                                        const long long* __restrict__ cols,
                                                        const float* __restrict__ vals,
                                                        const float* __restrict__ hidden,
                                                        float* __restrict__ support) {
  long long idx = (long long)blockIdx.x * blockDim.x + threadIdx.x;
  long long e  = idx >> 4;
  int       ch = (int)(idx & 15);
  if (e >= N_EDGES) return;
  long long r = rows[e];
  long long c = cols[e];
  float v = vals[e];
  float contrib = v * hidden[c * N_CLASSES + ch];
  atomic_add_f32_dev(&support[r * N_CLASSES + ch], contrib);
}

// ---------------------------------------------------------------------------
// Kernel 3: ragged pooling. Graph g owns nodes [ed_idx[g-1], ed_idx[g]);
// nodes >= ed_idx[G-1] are dropped (searchsorted 'right' semantics).
// One thread per (graph, channel).
// ---------------------------------------------------------------------------
__global__ __launch_bounds__(256) void gcn_pool(const long long* __restrict__ ed_idx,
                                                const float* __restrict__ support,
                                                float* __restrict__ out) {
  int tid = blockIdx.x * blockDim.x + threadIdx.x;
  if (tid >= N_GRAPHS * N_CLASSES) return;
  int g  = tid >> 4;
  int ch = tid & 15;
  long long start = (g == 0) ? 0 : ed_idx[g - 1];
  long long end   = ed_idx[g];
  float acc = 0.0f;
  for (long long n = start; n < end; ++n)
    acc += support[n * N_CLASSES + ch];
  out[tid] = acc;
}

// ---------------------------------------------------------------------------
extern "C" void kernel_launch(void* const* d_in, const int* in_sizes, int n_in,
                              void* d_out, int out_size, void* d_ws, size_t ws_size,
                              hipStream_t stream) {
  const float*     x      = (const float*)d_in[0];
  const long long* ed_idx = (const long long*)d_in[1];
  const long long* rows   = (const long long*)d_in[2];
  const long long* cols   = (const long long*)d_in[3];
  const float*     vals   = (const float*)d_in[4];
  const float*     W      = (const float*)d_in[5];
  const float*     bias   = (const float*)d_in[6];

  float* hidden  = (float*)d_ws;                              // N_NODES*16 f32 = 6.4 MB
  float* support = hidden + (size_t)N_NODES * N_CLASSES;      // N_NODES*16 f32 = 6.4 MB
  float* out     = (float*)d_out;

  const long long support_elems = (long long)N_NODES * N_CLASSES;

  // 0) zero the scatter target
  {
    int blocks = (int)((support_elems + 255) / 256);
    gcn_zero<<<blocks, 256, 0, stream>>>(support, support_elems);
  }
  // 1) hidden = x @ W + b   (WMMA f32 16x16x4)
  {
    int blocks = (N_NODES + 127) / 128;   // 128 rows per block (8 waves x 16 rows)
    gcn_gemm_bias<<<blocks, 256, 0, stream>>>(x, W, bias, hidden);
  }
  // 2) edge scatter with native fp32 atomics
  {
    long long threads = (long long)N_EDGES * N_CLASSES;
    int blocks = (int)((threads + 255) / 256);
    gcn_edge_scatter<<<blocks, 256, 0, stream>>>(rows, cols, vals, hidden, support);
  }
  // 3) ragged pool into d_out
  {
    int blocks = (N_GRAPHS * N_CLASSES + 255) / 256;
    gcn_pool<<<blocks, 256, 0, stream>>>(ed_idx, support, out);
  }
}